// LightGCN_9423158248257
// MI455X (gfx1250) — compile-verified
//
#include <hip/hip_runtime.h>
#include <math.h>

typedef __attribute__((ext_vector_type(2))) float v2f;
typedef __attribute__((ext_vector_type(8))) float v8f;

#define NUM_U 200000
#define NUM_I 100000
#define NUM_N 300000   // U + I
#define NUM_E 1000000
#define DIM   64

// ---------------------------------------------------------------------------
// 1) Edge MLP: w = clamp(sigmoid(relu(attr@W1+b1)@W2+b2), 1e-6)
// ---------------------------------------------------------------------------
__global__ __launch_bounds__(256) void edge_mlp_kernel(
    const float* __restrict__ attr, const float* __restrict__ W1,
    const float* __restrict__ b1, const float* __restrict__ W2,
    const float* __restrict__ b2, float* __restrict__ w_out, int E)
{
    __shared__ float sW1[8 * 32];
    __shared__ float sb1[32];
    __shared__ float sW2[32];
    __shared__ float sb2;
    int t = threadIdx.x;
    if (t < 256) sW1[t] = W1[t];
    if (t < 32) { sb1[t] = b1[t]; sW2[t] = W2[t]; }
    if (t == 0) sb2 = b2[0];
    __syncthreads();

    int e = blockIdx.x * 256 + t;
    if (e >= E) return;
    float a[8];
#pragma unroll
    for (int k = 0; k < 8; ++k) a[k] = attr[e * 8 + k];
    float acc = sb2;
#pragma unroll
    for (int j = 0; j < 32; ++j) {
        float h = sb1[j];
#pragma unroll
        for (int k = 0; k < 8; ++k) h = fmaf(a[k], sW1[k * 32 + j], h);
        acc = fmaf(fmaxf(h, 0.0f), sW2[j], acc);
    }
    float w = 1.0f / (1.0f + __expf(-acc));
    w_out[e] = fmaxf(w, 1e-6f);
}

// ---------------------------------------------------------------------------
// 2) Degree: deg = 1 (self loop) + incident edge weights; then dis = rsqrt(deg)
// ---------------------------------------------------------------------------
__global__ void deg_init_kernel(float* __restrict__ deg, int n)
{
    int i = blockIdx.x * blockDim.x + threadIdx.x;
    if (i < n) deg[i] = 1.0f;
}

__global__ void deg_acc_kernel(const int* __restrict__ esrc, const int* __restrict__ edst,
                               const float* __restrict__ w, float* __restrict__ deg, int E)
{
    int e = blockIdx.x * blockDim.x + threadIdx.x;
    if (e >= E) return;
    float we = w[e];
    atomicAdd(&deg[esrc[e]], we);
    atomicAdd(&deg[NUM_U + edst[e]], we);
}

__global__ void dis_kernel(float* __restrict__ deg, int n)
{
    int i = blockIdx.x * blockDim.x + threadIdx.x;
    if (i >= n) return;
    float d = deg[i];
    deg[i] = (d > 0.0f) ? rsqrtf(d) : 0.0f;
}

// ---------------------------------------------------------------------------
// 3a) Users: x[u] = l2norm(user_w[u]); one wave32 per user, 2 dwords/lane
// ---------------------------------------------------------------------------
__global__ __launch_bounds__(256) void user_norm_kernel(
    const float* __restrict__ uw, float* __restrict__ x)
{
    int gid = blockIdx.x * blockDim.x + threadIdx.x;
    int u = gid >> 5, lane = gid & 31;
    if (u >= NUM_U) return;
    size_t o = (size_t)u * DIM;
    float a = uw[o + lane], b = uw[o + 32 + lane];
    float s = a * a + b * b;
#pragma unroll
    for (int off = 16; off; off >>= 1) s += __shfl_xor(s, off, 32);
    float inv = 1.0f / fmaxf(sqrtf(s), 1e-12f);
    x[o + lane] = a * inv;
    x[o + 32 + lane] = b * inv;
}

// ---------------------------------------------------------------------------
// 3b) Items: x[U+i] = l2norm([audio, artist+album] @ Wp + bp)
//     One wave per 16 items; fp32 WMMA 16x16x4, 4 N-tiles, K=128 in steps of 4.
//     A layout (16x4 f32): lane m=L&15; lower half VGPR0/1 = K,K+1; upper = K+2,K+3
//     B layout mirrored: row K striped across lanes 0-15 (N = lane&15).
// ---------------------------------------------------------------------------
__global__ __launch_bounds__(32) void item_proj_kernel(
    const float* __restrict__ audio, const int* __restrict__ aid,
    const int* __restrict__ alid, const float* __restrict__ artist_w,
    const float* __restrict__ album_w, const float* __restrict__ Wp,
    const float* __restrict__ bp, float* __restrict__ x)
{
    __shared__ float lA[16][132];   // 16 items x 128 features (padded)
    __shared__ float lO[16][64];
    __shared__ float linv[16];

    const int lane = threadIdx.x;
    const int base = blockIdx.x * 16;

    // Stage the 16x128 input tile: [audio | artist_w[aid] + album_w[alid]]
    for (int idx = lane; idx < 16 * 128; idx += 32) {
        int r = idx >> 7, k = idx & 127;
        int it = base + r;
        float v;
        if (k < 64) {
            v = audio[(size_t)it * 64 + k];
        } else {
            int kk = k - 64;
            v = artist_w[(size_t)aid[it] * 64 + kk] + album_w[(size_t)alid[it] * 64 + kk];
        }
        lA[r][k] = v;
    }
    __syncthreads();

    v8f c0 = {}, c1 = {}, c2 = {}, c3 = {};
    const int m = lane & 15;
    const int khalf = (lane >> 4) << 1;   // 0 for lanes 0-15, 2 for lanes 16-31

    for (int k0 = 0; k0 < 128; k0 += 4) {
        int kb = k0 + khalf;
        v2f a;
        a.x = lA[m][kb];
        a.y = lA[m][kb + 1];
        const float* w0 = Wp + (size_t)kb * 64;
        const float* w1 = Wp + (size_t)(kb + 1) * 64;
        v2f b0, b1, b2, b3;
        b0.x = w0[m];       b0.y = w1[m];
        b1.x = w0[16 + m];  b1.y = w1[16 + m];
        b2.x = w0[32 + m];  b2.y = w1[32 + m];
        b3.x = w0[48 + m];  b3.y = w1[48 + m];
        c0 = __builtin_amdgcn_wmma_f32_16x16x4_f32(false, a, false, b0, (short)0, c0, false, false);
        c1 = __builtin_amdgcn_wmma_f32_16x16x4_f32(false, a, false, b1, (short)0, c1, false, false);
        c2 = __builtin_amdgcn_wmma_f32_16x16x4_f32(false, a, false, b2, (short)0, c2, false, false);
        c3 = __builtin_amdgcn_wmma_f32_16x16x4_f32(false, a, false, b3, (short)0, c3, false, false);
    }

    // Spill D tiles to LDS with bias. C/D layout: VGPR v holds M = v + 8*(lane>=16),
    // N = lane&15 within each 16-wide tile.
    const int mbase = (lane >> 4) << 3;   // 0 or 8
#pragma unroll
    for (int vi = 0; vi < 8; ++vi) {
        int row = mbase + vi;
        lO[row][m]      = c0[vi] + bp[m];
        lO[row][16 + m] = c1[vi] + bp[16 + m];
        lO[row][32 + m] = c2[vi] + bp[32 + m];
        lO[row][48 + m] = c3[vi] + bp[48 + m];
    }
    __syncthreads();

    if (lane < 16) {
        float s = 0.0f;
#pragma unroll
        for (int k = 0; k < 64; ++k) { float t = lO[lane][k]; s += t * t; }
        linv[lane] = 1.0f / fmaxf(sqrtf(s), 1e-12f);
    }
    __syncthreads();

    for (int idx = lane; idx < 16 * 64; idx += 32) {
        int r = idx >> 6, k = idx & 63;
        x[(size_t)(NUM_U + base + r) * DIM + k] = lO[r][k] * linv[r];
    }
}

// ---------------------------------------------------------------------------
// 4) Propagation helpers
// ---------------------------------------------------------------------------
__global__ void copy_kernel(const float* __restrict__ src, float* __restrict__ dst, int n)
{
    int i = blockIdx.x * blockDim.x + threadIdx.x;
    if (i < n) dst[i] = src[i];
}

// Self-loop term: y[n] = dis[n]^2 * x[n]
__global__ void prop_self_kernel(const float* __restrict__ x, const float* __restrict__ dis,
                                 float* __restrict__ y, int total)
{
    int i = blockIdx.x * blockDim.x + threadIdx.x;
    if (i >= total) return;
    float d = dis[i >> 6];
    y[i] = d * d * x[i];
}

// Both directions of one interaction edge per wave32; 2 dwords per lane.
__global__ __launch_bounds__(256) void prop_edge_kernel(
    const int* __restrict__ esrc, const int* __restrict__ edst,
    const float* __restrict__ w, const float* __restrict__ dis,
    const float* __restrict__ x, float* __restrict__ y, int E)
{
    int gid = blockIdx.x * 256 + threadIdx.x;
    int e = gid >> 5, lane = gid & 31;
    if (e >= E) return;
    int s = esrc[e];
    int d = NUM_U + edst[e];
    float nrm = dis[s] * w[e] * dis[d];
    size_t so = (size_t)s * DIM, dof = (size_t)d * DIM;
    float xs0 = x[so + lane],  xs1 = x[so + 32 + lane];
    float xd0 = x[dof + lane], xd1 = x[dof + 32 + lane];
    atomicAdd(&y[dof + lane],      nrm * xs0);
    atomicAdd(&y[dof + 32 + lane], nrm * xs1);
    atomicAdd(&y[so + lane],       nrm * xd0);
    atomicAdd(&y[so + 32 + lane],  nrm * xd1);
}

__global__ void acc_add_kernel(const float* __restrict__ y, float* __restrict__ acc, int n)
{
    int i = blockIdx.x * blockDim.x + threadIdx.x;
    if (i < n) acc[i] += y[i];
}

// ---------------------------------------------------------------------------
// 5) Finalize: out = l2norm(acc / (L+1)); in place, one wave32 per node.
// ---------------------------------------------------------------------------
__global__ __launch_bounds__(256) void final_norm_kernel(float* __restrict__ out)
{
    int gid = blockIdx.x * blockDim.x + threadIdx.x;
    int node = gid >> 5, lane = gid & 31;
    if (node >= NUM_N) return;
    size_t o = (size_t)node * DIM;
    float a = out[o + lane] * 0.25f;
    float b = out[o + 32 + lane] * 0.25f;
    float s = a * a + b * b;
#pragma unroll
    for (int off = 16; off; off >>= 1) s += __shfl_xor(s, off, 32);
    float inv = 1.0f / fmaxf(sqrtf(s), 1e-12f);
    out[o + lane] = a * inv;
    out[o + 32 + lane] = b * inv;
}

// ---------------------------------------------------------------------------
extern "C" void kernel_launch(void* const* d_in, const int* in_sizes, int n_in,
                              void* d_out, int out_size, void* d_ws, size_t ws_size,
                              hipStream_t stream)
{
    const int*   edge_src  = (const int*)  d_in[0];
    const int*   edge_dst  = (const int*)  d_in[1];
    const float* edge_attr = (const float*)d_in[2];
    const float* user_w    = (const float*)d_in[3];
    const float* artist_w  = (const float*)d_in[4];
    const float* album_w   = (const float*)d_in[5];
    const float* item_audio= (const float*)d_in[6];
    const int*   artist_ids= (const int*)  d_in[7];
    const int*   album_ids = (const int*)  d_in[8];
    const float* Wp        = (const float*)d_in[9];
    const float* bp        = (const float*)d_in[10];
    const float* W1        = (const float*)d_in[11];
    const float* b1        = (const float*)d_in[12];
    const float* W2        = (const float*)d_in[13];
    const float* b2        = (const float*)d_in[14];

    float* acc = (float*)d_out;                      // [N, 64] accumulator, finalized in place

    float* ws  = (float*)d_ws;
    float* w_e = ws;                                 // E
    float* dis = w_e + NUM_E;                        // N   (deg -> dis in place)
    float* xa  = dis + NUM_N;                        // N*64
    float* xb  = xa + (size_t)NUM_N * DIM;           // N*64

    const int total = NUM_N * DIM;

    // 1) edge weights
    edge_mlp_kernel<<<(NUM_E + 255) / 256, 256, 0, stream>>>(edge_attr, W1, b1, W2, b2, w_e, NUM_E);

    // 2) degree -> dis
    deg_init_kernel<<<(NUM_N + 255) / 256, 256, 0, stream>>>(dis, NUM_N);
    deg_acc_kernel<<<(NUM_E + 255) / 256, 256, 0, stream>>>(edge_src, edge_dst, w_e, dis, NUM_E);
    dis_kernel<<<(NUM_N + 255) / 256, 256, 0, stream>>>(dis, NUM_N);

    // 3) initial features into xa
    user_norm_kernel<<<(NUM_U * 32 + 255) / 256, 256, 0, stream>>>(user_w, xa);
    item_proj_kernel<<<NUM_I / 16, 32, 0, stream>>>(item_audio, artist_ids, album_ids,
                                                    artist_w, album_w, Wp, bp, xa);

    // 4) acc = x; 3 rounds of propagation with ping-pong buffers
    copy_kernel<<<(total + 255) / 256, 256, 0, stream>>>(xa, acc, total);

    float* xcur = xa;
    float* xnext = xb;
    for (int l = 0; l < 3; ++l) {
        prop_self_kernel<<<(total + 255) / 256, 256, 0, stream>>>(xcur, dis, xnext, total);
        prop_edge_kernel<<<(NUM_E * 32 + 255) / 256, 256, 0, stream>>>(edge_src, edge_dst, w_e,
                                                                       dis, xcur, xnext, NUM_E);
        acc_add_kernel<<<(total + 255) / 256, 256, 0, stream>>>(xnext, acc, total);
        float* t = xcur; xcur = xnext; xnext = t;
    }

    // 5) out = l2norm(acc / 4)
    final_norm_kernel<<<(NUM_N * 32 + 255) / 256, 256, 0, stream>>>(acc);
}